// HistogramLoss_90958817395096
// MI455X (gfx1250) — compile-verified
//
#include <hip/hip_runtime.h>

// Histogram-matching loss, MI455X (gfx1250, wave32).
// Pass 1: streaming per-block LDS histograms (counts u32, sum-of-values as u64
//         fixed point, 2^20 scale -> exact & deterministic), merged with global
//         integer atomics.
// Pass 2: single block: f32 CDFs, searchsorted transfer table, per-bin
//         contributions, then a v_wmma_f32_16x16x4_f32 chain reduces the 768
//         contributions on the matrix pipe.

#define H 2048
#define HH (H * H)          // 4194304 pixels
#define N4 (HH / 4)         // 1048576 float4s per channel
#define NBINS 256
#define NCH 3
#define TBINS (NCH * NBINS) // 768
#define FIXSCALE 1048576.0f // 2^20 fixed-point scale for sum(v)

typedef float v2f __attribute__((ext_vector_type(2)));
typedef float v8f __attribute__((ext_vector_type(8)));

__global__ void zero_ws_kernel(unsigned long long* __restrict__ ws) {
    int i = blockIdx.x * blockDim.x + threadIdx.x;
    if (i < 1536) ws[i] = 0ULL;   // 12288 bytes of accumulators
}

__global__ __launch_bounds__(256) void hist_kernel(
    const float4* __restrict__ in, const float4* __restrict__ tar,
    const float4* __restrict__ msrc, const float4* __restrict__ mtar,
    unsigned* __restrict__ g_cnt_src, unsigned* __restrict__ g_cnt_tar,
    unsigned long long* __restrict__ g_sumv)
{
    __shared__ unsigned s_cnt_src[TBINS];
    __shared__ unsigned s_cnt_tar[TBINS];
    __shared__ unsigned long long s_sumv[TBINS];
    for (int i = threadIdx.x; i < TBINS; i += 256) {
        s_cnt_src[i] = 0u; s_cnt_tar[i] = 0u; s_sumv[i] = 0ULL;
    }
    __syncthreads();

    const int stride = gridDim.x * blockDim.x;
    for (int p = blockIdx.x * blockDim.x + threadIdx.x; p < N4; p += stride) {
        float4 ms = msrc[p];
        float4 mt = mtar[p];
        if (p + stride < N4) {  // gfx1250 global_prefetch_b8 for next tile
            __builtin_prefetch(&msrc[p + stride], 0, 1);
            __builtin_prefetch(&mtar[p + stride], 0, 1);
        }
        float msv[4] = {ms.x, ms.y, ms.z, ms.w};
        float mtv[4] = {mt.x, mt.y, mt.z, mt.w};
        #pragma unroll
        for (int c = 0; c < NCH; ++c) {
            float4 iv = in[c * N4 + p];
            float4 tv = tar[c * N4 + p];
            float ivv[4] = {iv.x, iv.y, iv.z, iv.w};
            float tvv[4] = {tv.x, tv.y, tv.z, tv.w};
            #pragma unroll
            for (int j = 0; j < 4; ++j) {
                if (msv[j] != 0.0f) {
                    float v = fminf(fmaxf((ivv[j] + 1.0f) * 0.5f, 0.0f), 1.0f) * 255.0f;
                    int b = (int)v;                       // v in [0,255] -> trunc == ref clip
                    atomicAdd(&s_cnt_src[c * NBINS + b], 1u);
                    atomicAdd(&s_sumv[c * NBINS + b],
                              (unsigned long long)(v * FIXSCALE));
                }
                if (mtv[j] != 0.0f) {
                    float v = fminf(fmaxf((tvv[j] + 1.0f) * 0.5f, 0.0f), 1.0f) * 255.0f;
                    int b = (int)v;
                    atomicAdd(&s_cnt_tar[c * NBINS + b], 1u);
                }
            }
        }
    }
    __syncthreads();
    for (int i = threadIdx.x; i < TBINS; i += 256) {
        if (s_cnt_src[i]) atomicAdd(&g_cnt_src[i], s_cnt_src[i]);
        if (s_cnt_tar[i]) atomicAdd(&g_cnt_tar[i], s_cnt_tar[i]);
        if (s_sumv[i])    atomicAdd(&g_sumv[i],    s_sumv[i]);
    }
}

__global__ __launch_bounds__(256) void finalize_kernel(
    const unsigned* __restrict__ g_cnt_src, const unsigned* __restrict__ g_cnt_tar,
    const unsigned long long* __restrict__ g_sumv, float* __restrict__ out)
{
    __shared__ float cdf_dst[TBINS];
    __shared__ float cdf_ref[TBINS];
    __shared__ float contrib[TBINS];
    __shared__ float red[16];
    int tid = threadIdx.x;

    // Sequential f32 cumsum per channel (matches jnp.cumsum semantics).
    if (tid < NCH) {
        float ad = 0.0f, ar = 0.0f;
        const float invK = 1.0f / 1048576.0f;   // both hist sums are exactly 2^20
        for (int i = 0; i < NBINS; ++i) {
            ad += (float)g_cnt_src[tid * NBINS + i] * invK;
            ar += (float)g_cnt_tar[tid * NBINS + i] * invK;
            cdf_dst[tid * NBINS + i] = ad;
            cdf_ref[tid * NBINS + i] = ar;
        }
    }
    __syncthreads();

    // Transfer table + per-bin |v - t| contribution (exact: sign uniform per bin).
    for (int e = tid; e < TBINS; e += 256) {
        int c = e >> 8;
        int i = e & 255;
        float d = cdf_dst[e];
        const float* rc = &cdf_ref[c * NBINS];
        int j0 = 0;                                  // searchsorted 'left'
        for (int k = 0; k < NBINS; ++k) j0 += (rc[k] < d) ? 1 : 0;
        int j = j0 < 1 ? 1 : (j0 > NBINS - 1 ? NBINS - 1 : j0);
        bool valid = (rc[j - 1] <= d) && (d <= rc[j]);
        int t = valid ? j : i;
        if (i == NBINS - 1) t = NBINS - 1;
        float cnt = (float)g_cnt_src[e];
        float sv  = (float)((double)g_sumv[e] * (1.0 / 1048576.0));
        float tt  = (float)t;
        contrib[e] = (t <= i) ? (sv - cnt * tt) : (cnt * tt - sv);
    }
    __syncthreads();

    // Reduce 768 contributions on the matrix pipe: A = ones(16x4) so
    // D[0][n] = column sums of B; accumulate 12 chained WMMAs, then sum row 0.
    if (tid < 32) {                 // whole wave0: EXEC all-ones for WMMA
        v8f acc = {};
        v2f a; a[0] = 1.0f; a[1] = 1.0f;
        #pragma unroll
        for (int k = 0; k < 12; ++k) {
            v2f b;
            b[0] = contrib[k * 64 + tid];
            b[1] = contrib[k * 64 + 32 + tid];
            acc = __builtin_amdgcn_wmma_f32_16x16x4_f32(
                false, a, false, b, (short)0, acc, false, false);
        }
        if (tid < 16) red[tid] = acc[0];   // D row 0 (M=0), N = lane
    }
    __syncthreads();
    if (tid == 0) {
        float s = 0.0f;
        for (int n = 0; n < 16; ++n) s += red[n];
        out[0] = s / (float)(NCH * HH);    // mean over full 3xHxH tensor
    }
}

extern "C" void kernel_launch(void* const* d_in, const int* in_sizes, int n_in,
                              void* d_out, int out_size, void* d_ws, size_t ws_size,
                              hipStream_t stream) {
    (void)in_sizes; (void)n_in; (void)out_size; (void)ws_size;
    const float4* in = (const float4*)d_in[0];
    const float4* tr = (const float4*)d_in[1];
    const float4* ms = (const float4*)d_in[2];
    const float4* mt = (const float4*)d_in[3];

    unsigned* cnt_src = (unsigned*)d_ws;                       // 768 u32
    unsigned* cnt_tar = cnt_src + TBINS;                       // 768 u32
    unsigned long long* sumv =
        (unsigned long long*)((char*)d_ws + 2 * TBINS * sizeof(unsigned)); // 768 u64

    zero_ws_kernel<<<6, 256, 0, stream>>>((unsigned long long*)d_ws);
    hist_kernel<<<1024, 256, 0, stream>>>(in, tr, ms, mt, cnt_src, cnt_tar, sumv);
    finalize_kernel<<<1, 256, 0, stream>>>(cnt_src, cnt_tar, sumv, (float*)d_out);
}